// RecurrentLinearAttention_49641232007296
// MI455X (gfx1250) — compile-verified
//
#include <hip/hip_runtime.h>

typedef __attribute__((ext_vector_type(2))) float v2f;
typedef __attribute__((ext_vector_type(8))) float v8f;

#define RLA_EPS 1e-6f

__device__ __forceinline__ float feat(float x) {
    // elu(x)+1 == x+1 for x>0, exp(x) for x<=0
    return x > 0.0f ? x + 1.0f : __expf(x);
}

// Shapes: N=2048, H=8, D=64, M=64. One wave (32 lanes) per (n,h) pair.
// grid = N blocks of 256 threads (8 waves); wave w handles head h=w.
__global__ __launch_bounds__(256) void rla_step_kernel(
    const float* __restrict__ q, const float* __restrict__ k,
    const float* __restrict__ vv, const float* __restrict__ Si,
    const float* __restrict__ Zi,
    float* __restrict__ outV, float* __restrict__ outSi,
    float* __restrict__ outZi, int H)
{
    constexpr int D = 64, M = 64;
    __shared__ float sQ[8][64];
    __shared__ float sK[8][64];
    __shared__ float sV[8][64];

    const int tid  = threadIdx.x;
    const int h    = tid >> 5;
    const int lane = tid & 31;
    const int n    = blockIdx.x;
    const int nh   = n * H + h;

    const size_t vecBase = (size_t)nh * D;              // q,k,v,Zi,outZi,outV bases
    const size_t matBase = (size_t)nh * (size_t)(D * M);

    // Each lane loads 2 of the 64 vector elements.
    float q0 = q[vecBase + lane],  q1 = q[vecBase + lane + 32];
    float k0 = k[vecBase + lane],  k1 = k[vecBase + lane + 32];
    float v0 = vv[vecBase + lane], v1 = vv[vecBase + lane + 32];
    float z0 = Zi[vecBase + lane], z1 = Zi[vecBase + lane + 32];

    float Q0 = feat(q0), Q1 = feat(q1);
    float K0 = feat(k0), K1 = feat(k1);
    float zn0 = z0 + K0, zn1 = z1 + K1;

    outZi[vecBase + lane]      = zn0;
    outZi[vecBase + lane + 32] = zn1;

    sQ[h][lane] = Q0; sQ[h][lane + 32] = Q1;
    sK[h][lane] = K0; sK[h][lane + 32] = K1;
    sV[h][lane] = v0; sV[h][lane + 32] = v1;

    // Z = 1/(dot(Q, Zi_new) + eps): wave-wide reduction (wave32).
    float part = Q0 * zn0 + Q1 * zn1;
    #pragma unroll
    for (int off = 16; off >= 1; off >>= 1)
        part += __shfl_xor(part, off, 32);
    const float Zinv = 1.0f / (part + RLA_EPS);

    const int hl  = lane >> 4;   // lane half: 0 -> rows r, 1 -> rows r+8
    const int col = lane & 15;   // column within 16-wide tile

    // Hoist A/B source values: unconditional LDS reads (lanes 16-31 read the
    // same addresses as 0-15 -> broadcast, no bank conflict), then select to
    // zero in the upper half (v_cndmask, no EXEC divergence around ds loads).
    float kA[4], vB[4];
    #pragma unroll
    for (int b = 0; b < 4; ++b) {
        float kvRaw = sK[h][b * 16 + col];
        float vvRaw = sV[h][b * 16 + col];
        kA[b] = (hl == 0) ? kvRaw : 0.0f;
        vB[b] = (hl == 0) ? vvRaw : 0.0f;
    }

    // Per-lane base pointers; every tile access below is an immediate offset.
    const float* cbase = Si    + matBase + (size_t)(hl * 8) * M + col;
    float*       obase = outSi + matBase + (size_t)(hl * 8) * M + col;

    // Si_new = Si + K (outer) V via V_WMMA_F32_16X16X4_F32 rank-1 update,
    // fused with V_out[m] = Zinv * sum_d Q[d]*Si_new[d][m].
    #pragma unroll
    for (int mb = 0; mb < 4; ++mb) {
        const int m0 = mb * 16;
        // B (4x16): row K=0 = V[m0..m0+15] in lanes 0-15 of VGPR0; rest zero.
        v2f B;
        B.x = vB[mb];
        B.y = 0.0f;

        float acc = 0.0f;
        #pragma unroll
        for (int db = 0; db < 4; ++db) {
            const int d0 = db * 16;
            // A (16x4): col K=0 = K[d0..d0+15] in lanes 0-15 of VGPR0; rest zero.
            v2f A;
            A.x = kA[db];
            A.y = 0.0f;

            // Load C tile in the WMMA C/D layout:
            // VGPR r: lanes 0-15 = Si[d0+r][m0+col], lanes 16-31 = Si[d0+r+8][m0+col]
            v8f C;
            #pragma unroll
            for (int r = 0; r < 8; ++r)
                C[r] = __builtin_nontemporal_load(cbase + (d0 + r) * M + m0);

            v8f Dt = __builtin_amdgcn_wmma_f32_16x16x4_f32(
                false, A, false, B, (short)0, C, false, false);

            // Stream out Si_new tile (non-temporal: 256MB, touched once).
            #pragma unroll
            for (int r = 0; r < 8; ++r)
                __builtin_nontemporal_store(Dt[r], obase + (d0 + r) * M + m0);

            // Partial matvec on the register tile: this lane owns rows
            // d0 + hl*8 + r of column m0+col.
            const float* qrow = &sQ[h][d0 + hl * 8];
            #pragma unroll
            for (int r = 0; r < 8; ++r)
                acc += qrow[r] * Dt[r];
        }
        // Merge the two lane-halves (rows 0-7,16-23,... vs 8-15,24-31,...).
        acc += __shfl_xor(acc, 16, 32);
        if (hl == 0)
            outV[(size_t)nh * M + m0 + col] = acc * Zinv;
    }
}

extern "C" void kernel_launch(void* const* d_in, const int* in_sizes, int n_in,
                              void* d_out, int out_size, void* d_ws, size_t ws_size,
                              hipStream_t stream) {
    const float* q  = (const float*)d_in[0];
    const float* k  = (const float*)d_in[1];
    const float* v  = (const float*)d_in[2];
    const float* Si = (const float*)d_in[3];
    const float* Zi = (const float*)d_in[4];

    const int H = 8, D = 64, M = 64;
    const int N = in_sizes[0] / (H * D);   // 2048 per reference

    float* outV  = (float*)d_out;                       // (N,H,M)
    float* outSi = outV  + (size_t)N * H * M;           // (N,H,D,M)
    float* outZi = outSi + (size_t)N * H * D * M;       // (N,H,D)

    rla_step_kernel<<<N, 256, 0, stream>>>(q, k, v, Si, Zi, outV, outSi, outZi, H);
}